// Simplify_S6_feat2offset_80023830659194
// MI455X (gfx1250) — compile-verified
//
#include <hip/hip_runtime.h>
#include <hip/hip_bf16.h>
#include <cstdint>

// ---------------- problem constants (from the reference) ----------------
#define B_SZ   16
#define D_IN   512
#define SEQ    4096
#define NST    16            // D_STATE
#define RNK    32            // DT_RANK
#define E_DIM  64            // RNK + 2*NST
#define NCH    16            // chunks over SEQ
#define CHL    (SEQ / NCH)   // 256 steps per chunk
#define TL     32            // LDS staging tile (timesteps)

typedef float v2f __attribute__((ext_vector_type(2)));
typedef float v8f __attribute__((ext_vector_type(8)));

// ---------------------------------------------------------------------------
// CDNA5 async global->LDS copy (bypasses VGPRs, tracked by ASYNCcnt).
// laddr = LDS byte address (low 32 bits of a generic pointer to __shared__).
// ---------------------------------------------------------------------------
__device__ __forceinline__ void async_g2lds_b128(uint32_t laddr, const float* gaddr) {
  asm volatile("global_load_async_to_lds_b128 %0, %1, off"
               :: "v"(laddr), "v"(gaddr) : "memory");
}
__device__ __forceinline__ void wait_asynccnt0() {
  asm volatile("s_wait_asynccnt 0x0" ::: "memory");
}

// ---------------------------------------------------------------------------
// Kernel 1: x_dbl[b,e,l] = sum_d x_proj_w[e,d] * x[b,d,l]
// Per batch: (64 x 512) @ (512 x 4096). WMMA f32 16x16x4.
// Block = 128 threads (4 waves). Block tile: M=64 (full E), N=64.
// ---------------------------------------------------------------------------
__global__ __launch_bounds__(128) void xproj_wmma(
    const float* __restrict__ x,        // (B, D, L)
    const float* __restrict__ w,        // (E, D)
    float* __restrict__ xdbl)           // (B, E, L)
{
  const int b    = blockIdx.y;
  const int n0   = blockIdx.x * 64;
  const int wave = threadIdx.x >> 5;
  const int lane = threadIdx.x & 31;
  const int m0   = wave * 16;
  const int lm   = lane & 15;
  const int kk   = (lane >> 4) * 2;     // 0 or 2

  const float* __restrict__ xb = x + (size_t)b * D_IN * SEQ;

  v8f acc0 = {}, acc1 = {}, acc2 = {}, acc3 = {};
  for (int k0 = 0; k0 < D_IN; k0 += 4) {
    v2f a;
    a.x = w[(m0 + lm) * D_IN + k0 + kk];
    a.y = w[(m0 + lm) * D_IN + k0 + kk + 1];

    const float* __restrict__ xr0 = xb + (size_t)(k0 + kk) * SEQ + n0 + lm;
    const float* __restrict__ xr1 = xr0 + SEQ;

    v2f b0; b0.x = xr0[ 0]; b0.y = xr1[ 0];
    v2f b1; b1.x = xr0[16]; b1.y = xr1[16];
    v2f b2; b2.x = xr0[32]; b2.y = xr1[32];
    v2f b3; b3.x = xr0[48]; b3.y = xr1[48];

    acc0 = __builtin_amdgcn_wmma_f32_16x16x4_f32(false, a, false, b0, (short)0, acc0, false, false);
    acc1 = __builtin_amdgcn_wmma_f32_16x16x4_f32(false, a, false, b1, (short)0, acc1, false, false);
    acc2 = __builtin_amdgcn_wmma_f32_16x16x4_f32(false, a, false, b2, (short)0, acc2, false, false);
    acc3 = __builtin_amdgcn_wmma_f32_16x16x4_f32(false, a, false, b3, (short)0, acc3, false, false);
  }

  float* __restrict__ outb = xdbl + (size_t)b * E_DIM * SEQ;
  const int rbase = m0 + (lane >> 4) * 8;
  v8f accs[4] = {acc0, acc1, acc2, acc3};
#pragma unroll
  for (int t = 0; t < 4; ++t) {
    const int col = n0 + t * 16 + lm;
#pragma unroll
    for (int v = 0; v < 8; ++v) {
      outb[(size_t)(rbase + v) * SEQ + col] = accs[t][v];
    }
  }
}

// ---------------------------------------------------------------------------
// Scan pass. 128 threads = 128 d-channels; one block = one (b, dblk, chunk).
// PASS1: accumulate Sdelta and h_partial (h_start = 0).
// PASS2: seed h from Hinit and emit y.
// dt_low/B/C tiles staged into LDS via async global->LDS B128 copies; at each
// step all lanes read the same LDS word (uniform l) -> broadcast reads.
// ---------------------------------------------------------------------------
template <bool PASS1>
__global__ __launch_bounds__(128) void scan_pass(
    const float* __restrict__ x,         // (B, D, L)
    const float* __restrict__ xdbl,      // (B, E, L): [0:32)=dt_low, [32:48)=B, [48:64)=C
    const float* __restrict__ dt_proj_w, // (D, R)
    const float* __restrict__ dt_proj_b, // (D,)
    const float* __restrict__ A_log,     // (D, N)
    float* __restrict__ Ssum,            // (B, NCH, D)         [pass1 out]
    float* __restrict__ Hpart,           // (B, NCH, D, N)      [pass1 out]
    const float* __restrict__ Hinit,     // (B, NCH, D, N)      [pass2 in]
    float* __restrict__ y)               // (B, D, L)           [pass2 out]
{
  const int chunk = blockIdx.x;
  const int d     = blockIdx.y * 128 + threadIdx.x;
  const int b     = blockIdx.z;
  const int tid   = threadIdx.x;
  const int lbeg  = chunk * CHL;

  __shared__ float s_all[E_DIM * TL];          // [e][l] : 8 KB
  const float* __restrict__ s_dt = s_all;                    // e in [0,32)
  const float* __restrict__ s_B  = s_all + RNK * TL;         // e in [32,48)
  const float* __restrict__ s_C  = s_all + (RNK + NST) * TL; // e in [48,64)

  // per-channel constants
  float An[NST];
#pragma unroll
  for (int n = 0; n < NST; ++n) An[n] = -__expf(A_log[d * NST + n]);
  float dtw[RNK];
#pragma unroll
  for (int r = 0; r < RNK; ++r) dtw[r] = dt_proj_w[d * RNK + r];
  const float bias = dt_proj_b[d];

  const size_t sum_base = ((size_t)b * NCH + chunk) * D_IN + d;

  float h[NST];
  if (PASS1) {
#pragma unroll
    for (int n = 0; n < NST; ++n) h[n] = 0.0f;
  } else {
#pragma unroll
    for (int n = 0; n < NST; ++n) h[n] = Hinit[sum_base * NST + n];
  }
  float Sdelta = 0.0f;

  const float* __restrict__ xrow = x + ((size_t)b * D_IN + d) * SEQ;
  float* __restrict__ yrow = PASS1 ? nullptr : (y + ((size_t)b * D_IN + d) * SEQ);
  const float* __restrict__ xdb = xdbl + (size_t)b * E_DIM * SEQ;

  // pass1 does not need the C rows
  constexpr int NROW   = PASS1 ? (RNK + NST) : E_DIM;
  constexpr int NSLOTS = NROW * (TL / 4);      // float4 slots per stage

  for (int l0 = lbeg; l0 < lbeg + CHL; l0 += TL) {
    // ---- async stage of dt_low/B(/C) tile into LDS (B128 per issue) ----
#pragma unroll
    for (int s = tid; s < NSLOTS; s += 128) {
      const int e  = s >> 3;                   // TL/4 == 8 slots per row
      const int l4 = (s & 7) * 4;
      const uint32_t laddr =
          (uint32_t)(uintptr_t)(s_all + e * TL + l4);
      async_g2lds_b128(laddr, xdb + (size_t)e * SEQ + l0 + l4);
    }
    wait_asynccnt0();
    __syncthreads();

    for (int lt = 0; lt < TL; lt += 4) {
      const float4 u4 = *reinterpret_cast<const float4*>(xrow + l0 + lt);
      float y4[4];
#pragma unroll
      for (int j = 0; j < 4; ++j) {
        const int ll = lt + j;
        const float u = (j == 0) ? u4.x : (j == 1) ? u4.y : (j == 2) ? u4.z : u4.w;

        // delta = softplus( dt_proj_w[d,:] . dt_low[b,:,l] + bias )
        float dpre = bias;
#pragma unroll
        for (int r = 0; r < RNK; ++r) dpre = fmaf(dtw[r], s_dt[r * TL + ll], dpre);
        const float delta = fmaxf(dpre, 0.0f) + log1pf(__expf(-fabsf(dpre)));

        const float du = delta * u;
        float yv = 0.0f;
#pragma unroll
        for (int n = 0; n < NST; ++n) {
          const float dA = __expf(delta * An[n]);
          h[n] = fmaf(dA, h[n], du * s_B[n * TL + ll]);
          if (!PASS1) yv = fmaf(h[n], s_C[n * TL + ll], yv);
        }
        if (PASS1) Sdelta += delta;
        y4[j] = yv;
      }
      if (!PASS1) {
        *reinterpret_cast<float4*>(yrow + l0 + lt) =
            make_float4(y4[0], y4[1], y4[2], y4[3]);
      }
    }
    __syncthreads();
  }

  if (PASS1) {
    Ssum[sum_base] = Sdelta;
#pragma unroll
    for (int n = 0; n < NST; ++n) Hpart[sum_base * NST + n] = h[n];
  }
}

// ---------------------------------------------------------------------------
// Kernel 3: sequential combine over chunks (parallel over B*D = 8192 lanes).
// h_{c+1} = exp(A * Ssum_c) * h_c + Hpart_c ; Hinit_c = h_c.
// ---------------------------------------------------------------------------
__global__ __launch_bounds__(256) void combine_chunks(
    const float* __restrict__ Ssum,   // (B, NCH, D)
    const float* __restrict__ Hpart,  // (B, NCH, D, N)
    float* __restrict__ Hinit,        // (B, NCH, D, N)
    const float* __restrict__ A_log)  // (D, N)
{
  const int idx = blockIdx.x * blockDim.x + threadIdx.x;  // over B*D
  if (idx >= B_SZ * D_IN) return;
  const int b = idx / D_IN;
  const int d = idx % D_IN;

  float An[NST];
#pragma unroll
  for (int n = 0; n < NST; ++n) An[n] = -__expf(A_log[d * NST + n]);

  float h[NST];
#pragma unroll
  for (int n = 0; n < NST; ++n) h[n] = 0.0f;

  for (int c = 0; c < NCH; ++c) {
    const size_t base = ((size_t)b * NCH + c) * D_IN + d;
#pragma unroll
    for (int n = 0; n < NST; ++n) Hinit[base * NST + n] = h[n];
    const float S = Ssum[base];
#pragma unroll
    for (int n = 0; n < NST; ++n)
      h[n] = fmaf(__expf(An[n] * S), h[n], Hpart[base * NST + n]);
  }
}

// ---------------------------------------------------------------------------
// Host-side launch
// ---------------------------------------------------------------------------
extern "C" void kernel_launch(void* const* d_in, const int* in_sizes, int n_in,
                              void* d_out, int out_size, void* d_ws, size_t ws_size,
                              hipStream_t stream) {
  const float* x         = (const float*)d_in[0];  // (16,512,4096)
  const float* x_proj_w  = (const float*)d_in[1];  // (64,512)
  const float* dt_proj_w = (const float*)d_in[2];  // (512,32)
  const float* dt_proj_b = (const float*)d_in[3];  // (512,)
  const float* A_log     = (const float*)d_in[4];  // (512,16)
  float* out = (float*)d_out;                      // (16,512,4096)

  // workspace layout (floats)
  float* ws    = (float*)d_ws;
  float* xdbl  = ws;                                      // B*E*L      = 4,194,304
  float* Ssum  = xdbl + (size_t)B_SZ * E_DIM * SEQ;       // B*NCH*D    =   131,072
  float* Hpart = Ssum + (size_t)B_SZ * NCH * D_IN;        // B*NCH*D*N  = 2,097,152
  float* Hinit = Hpart + (size_t)B_SZ * NCH * D_IN * NST; //            = 2,097,152
  // total ~34.1 MB

  // 1) x_dbl GEMM (WMMA f32 16x16x4)
  {
    dim3 grid(SEQ / 64, B_SZ);
    xproj_wmma<<<grid, 128, 0, stream>>>(x, x_proj_w, xdbl);
  }
  // 2) pass 1: chunk summaries (Sdelta, h_partial)
  {
    dim3 grid(NCH, D_IN / 128, B_SZ);
    scan_pass<true><<<grid, 128, 0, stream>>>(x, xdbl, dt_proj_w, dt_proj_b,
                                              A_log, Ssum, Hpart, nullptr, nullptr);
  }
  // 3) sequential chunk combine -> per-chunk initial states
  {
    dim3 grid((B_SZ * D_IN + 255) / 256);
    combine_chunks<<<grid, 256, 0, stream>>>(Ssum, Hpart, Hinit, A_log);
  }
  // 4) pass 2: full scan with correct initial states, emit y
  {
    dim3 grid(NCH, D_IN / 128, B_SZ);
    scan_pass<false><<<grid, 128, 0, stream>>>(x, xdbl, dt_proj_w, dt_proj_b,
                                               A_log, nullptr, nullptr, Hinit, out);
  }
}